// LightGCNConv_86337432584536
// MI455X (gfx1250) — compile-verified
//
#include <hip/hip_runtime.h>
#include <math.h>

#define EMB 64
#define EPS 1e-12f

typedef __attribute__((ext_vector_type(2))) float v2f;
typedef __attribute__((ext_vector_type(8))) float v8f;

// ---------------------------------------------------------------------------
// Kernel 1: zero the accumulator (harness poisons d_out; we must re-zero).
// ---------------------------------------------------------------------------
__global__ void zero_kernel(float4* __restrict__ out, int n4) {
  int i = blockIdx.x * blockDim.x + threadIdx.x;
  if (i < n4) out[i] = make_float4(0.f, 0.f, 0.f, 0.f);
}

// ---------------------------------------------------------------------------
// Kernel 2: edge scatter.  16 lanes per edge: each lane gathers one float4 of
// h[src] (full 256B row, coalesced), scales by w, and does 4 f32 atomics into
// out[dst].  Table + accumulator are L2-resident (18MB each << 192MB L2).
// Edge streams are prefetched ahead (global_prefetch_b8).
// ---------------------------------------------------------------------------
__global__ __launch_bounds__(256) void scatter_kernel(
    const float* __restrict__ h, const float* __restrict__ w,
    const int* __restrict__ src, const int* __restrict__ dst,
    float* __restrict__ out, int nE)
{
  const int gid    = blockIdx.x * blockDim.x + threadIdx.x;
  const int lane16 = gid & 15;
  const int estep  = (gridDim.x * blockDim.x) >> 4;

  for (int e = gid >> 4; e < nE; e += estep) {
    if (lane16 == 0) {
      int ep = e + estep * 4;
      if (ep < nE) {
        __builtin_prefetch(&src[ep], 0, 0);
        __builtin_prefetch(&dst[ep], 0, 0);
        __builtin_prefetch(&w[ep],   0, 0);
      }
    }
    const float wgt = w[e];
    const int   s   = src[e];
    const int   d   = dst[e];
    const float4 v  = ((const float4*)h)[s * (EMB / 4) + lane16];
    float* o = out + (size_t)d * EMB + lane16 * 4;
    atomicAdd(o + 0, v.x * wgt);
    atomicAdd(o + 1, v.y * wgt);
    atomicAdd(o + 2, v.z * wgt);
    atomicAdd(o + 3, v.w * wgt);
  }
}

// ---------------------------------------------------------------------------
// Kernel 3: row L2-normalize, in place.  One wave32 handles a 16-node tile.
// Squared row norms = diag(X · X^T); computed with chained
// V_WMMA_F32_16X16X4_F32.  Per the ISA VGPR layouts, the A(16x4) layout and
// the B(4x16)=A^T layout put identical values in identical lane/VGPR slots:
//   A: lane m (m=lane&15), VGPR0/1 = K = {0,1} (lanes 0-15) or {2,3} (16-31)
//   B: VGPR0/1 lanes0-15 = rows K=0/1 at N=lane; lanes16-31 = rows K=2/3
// so we pass the same v2f as both operands.  Diagonal extraction:
//   D[m][m] lives in c[m] of lane m (m<8) or c[m-8] of lane m+16 (m>=8).
// ---------------------------------------------------------------------------
__global__ __launch_bounds__(256) void normalize_kernel(float* __restrict__ out,
                                                        int nNodes)
{
  const int tid      = blockIdx.x * blockDim.x + threadIdx.x;
  const int lane     = threadIdx.x & 31;
  const int wlocal   = threadIdx.x >> 5;
  const int m        = lane & 15;
  const int half     = lane >> 4;          // 0: K/dims 0-31, 1: 32-63 role
  const int tileBase = (tid >> 5) * 16;
  const int node     = tileBase + m;
  const int nodeC    = node < nNodes ? node : nNodes - 1;   // clamp (no branch)
  const float2* row2 = (const float2*)(out + (size_t)nodeC * EMB);

#if __has_builtin(__builtin_amdgcn_wmma_f32_16x16x4_f32)
  __shared__ float diag[8][32][8];
  v8f c = {};
  #pragma unroll
  for (int kc = 0; kc < 16; ++kc) {
    // columns 4*kc + {0,1} (half==0) or {2,3} (half==1)  -> A and B layouts
    float2 t = row2[kc * 2 + half];
    v2f a; a.x = t.x; a.y = t.y;
    c = __builtin_amdgcn_wmma_f32_16x16x4_f32(
            /*neg_a=*/false, a, /*neg_b=*/false, a,
            /*c_mod=*/(short)0, c, /*reuse_a=*/false, /*reuse_b=*/false);
  }
  #pragma unroll
  for (int j = 0; j < 8; ++j) diag[wlocal][lane][j] = c[j];
  __syncthreads();
  const int   srcLane = (m < 8) ? m : (m + 16);
  const float n2      = diag[wlocal][srcLane][m & 7];
#else
  // VALU fallback: each lane sums squares of its 32 dims, pair-reduce.
  float partial = 0.f;
  #pragma unroll
  for (int j = 0; j < 16; ++j) {
    float2 t = row2[half * 16 + j];
    partial += t.x * t.x + t.y * t.y;
  }
  const float n2 = partial + __shfl_xor(partial, 16, 32);
#endif

  const float scale = 1.0f / fmaxf(sqrtf(n2), EPS);

  if (node < nNodes) {
    float4* r4 = (float4*)(out + (size_t)node * EMB + half * 32);
    #pragma unroll
    for (int j = 0; j < 8; ++j) {
      float4 v = r4[j];
      v.x *= scale; v.y *= scale; v.z *= scale; v.w *= scale;
      r4[j] = v;
    }
  }
}

// ---------------------------------------------------------------------------
extern "C" void kernel_launch(void* const* d_in, const int* in_sizes, int n_in,
                              void* d_out, int out_size, void* d_ws, size_t ws_size,
                              hipStream_t stream) {
  const float* emb = (const float*)d_in[0];   // [N, 64] f32
  const float* w   = (const float*)d_in[1];   // [E]     f32
  const int*   src = (const int*)d_in[2];     // [E]     i32
  const int*   dst = (const int*)d_in[3];     // [E]     i32
  float*       out = (float*)d_out;           // [N, 64] f32

  const int nNodes = in_sizes[0] / EMB;
  const int nE     = in_sizes[1];

  // 1) zero accumulator
  const int n4 = nNodes * (EMB / 4);
  zero_kernel<<<(n4 + 255) / 256, 256, 0, stream>>>((float4*)out, n4);

  // 2) gather-scale-scatter (grid-stride; 16 lanes per edge)
  scatter_kernel<<<4096, 256, 0, stream>>>(emb, w, src, dst, out, nE);

  // 3) in-place row normalize (one wave32 per 16-node tile)
  normalize_kernel<<<(nNodes + 127) / 128, 256, 0, stream>>>(out, nNodes);
}